// GraphSAGE_48842368090622
// MI455X (gfx1250) — compile-verified
//
#include <hip/hip_runtime.h>

typedef float v2f __attribute__((ext_vector_type(2)));
typedef float v8f __attribute__((ext_vector_type(8)));

#define KDIM 128      // both layers have din == 128
#define LDA  132      // padded LDS row stride (floats): 132 % 64 == 4 -> conflict-free column reads

// ---------------------------------------------------------------------------
// Dual-input GEMM: C[N, NT*16] = act( A1 @ W1 (+ A2 @ W2) + b1 (+ b2) )
// One wave computes one 16-row M-tile across all NT 16-col N-tiles.
// A-tile staged in LDS (coalesced float4), WMMA f32 16x16x4 over K=128.
// ---------------------------------------------------------------------------
template <int NT>
__global__ __launch_bounds__(128) void sage_gemm(
    const float* __restrict__ A1, const float* __restrict__ W1, const float* __restrict__ b1,
    const float* __restrict__ A2, const float* __restrict__ W2, const float* __restrict__ b2,
    float* __restrict__ C, int N, int relu)
{
    __shared__ float lds[4 * 16 * LDA];                 // 4 waves/block, 16x132 each
    const int lane  = threadIdx.x & 31;
    const int wave  = threadIdx.x >> 5;
    const int tileM = blockIdx.x * 4 + wave;
    const int row0  = tileM * 16;
    if (row0 >= N) return;                              // wave-uniform: EXEC stays all-1s
    float* sl = lds + wave * (16 * LDA);

    v8f acc[NT] = {};

    const int nlane = lane & 15;
    const int khalf = (lane >> 4) * 2;                  // lanes 0-15 -> K+0/1, 16-31 -> K+2/3

    const int dout = NT * 16;

    for (int pass = 0; pass < 2; ++pass) {
        const float* A = pass ? A2 : A1;
        const float* W = pass ? W2 : W1;
        if (pass && A2 == nullptr) break;

        // Stage 16x128 A block into LDS, coalesced: 512 float4s, 16 per lane.
        const float4* Ag = (const float4*)(A + (size_t)row0 * KDIM);
        #pragma unroll
        for (int it = 0; it < 16; ++it) {
            int f4 = it * 32 + lane;                    // 0..511
            int r  = f4 >> 5;                           // row within tile
            int c4 = f4 & 31;                           // float4 column
            float4 v = Ag[f4];
            *(float4*)(sl + r * LDA + c4 * 4) = v;
        }
        // same-wave LDS RAW: compiler inserts s_wait_dscnt

        #pragma unroll 4
        for (int ks = 0; ks < KDIM / 4; ++ks) {
            const int k0 = ks * 4 + khalf;
            v2f a;
            a.x = sl[nlane * LDA + k0];
            a.y = sl[nlane * LDA + k0 + 1];
            const float* Wk0 = W + (size_t)k0 * dout;   // row k0 of W
            const float* Wk1 = Wk0 + dout;              // row k0+1
            #pragma unroll
            for (int t = 0; t < NT; ++t) {
                v2f b;
                b.x = Wk0[t * 16 + nlane];
                b.y = Wk1[t * 16 + nlane];
                acc[t] = __builtin_amdgcn_wmma_f32_16x16x4_f32(
                    false, a, false, b, (short)0, acc[t], false, false);
            }
        }
    }

    // D layout: VGPR j -> row j (lanes 0-15) / row j+8 (lanes 16-31), col = nlane
    const int roff = (lane >> 4) * 8;
    #pragma unroll
    for (int t = 0; t < NT; ++t) {
        const int col = t * 16 + nlane;
        float bv = b1[col] + (b2 ? b2[col] : 0.0f);
        #pragma unroll
        for (int j = 0; j < 8; ++j) {
            float v = acc[t][j] + bv;
            if (relu) v = fmaxf(v, 0.0f);
            C[(size_t)(row0 + roff + j) * dout + col] = v;
        }
    }
}

// ---------------------------------------------------------------------------
// Edge scatter-max: one wave per edge. Lane handles 4 consecutive features.
// Values are relu'd (>= 0), so u32 atomic max == f32 max; agg pre-zeroed,
// which also implements maximum(agg, 0) for isolated nodes.
// ---------------------------------------------------------------------------
__global__ __launch_bounds__(256) void sage_scatter_max(
    const float* __restrict__ pooled, const int* __restrict__ src,
    const int* __restrict__ dst, unsigned int* __restrict__ agg, int E)
{
    const int gid  = blockIdx.x * blockDim.x + threadIdx.x;
    const int edge = gid >> 5;
    const int lane = gid & 31;
    if (edge >= E) return;
    const int s = src[edge];
    const int d = dst[edge];
    float4 v = *(const float4*)(pooled + (size_t)s * 128 + lane * 4);
    unsigned int* p = agg + (size_t)d * 128 + lane * 4;
    atomicMax(p + 0, __float_as_uint(v.x));
    atomicMax(p + 1, __float_as_uint(v.y));
    atomicMax(p + 2, __float_as_uint(v.z));
    atomicMax(p + 3, __float_as_uint(v.w));
}

__global__ __launch_bounds__(256) void zero_kernel(float4* __restrict__ p, int n4)
{
    int i = blockIdx.x * blockDim.x + threadIdx.x;
    int stride = gridDim.x * blockDim.x;
    float4 z = {0.0f, 0.0f, 0.0f, 0.0f};
    for (; i < n4; i += stride) p[i] = z;
}

extern "C" void kernel_launch(void* const* d_in, const int* in_sizes, int n_in,
                              void* d_out, int out_size, void* d_ws, size_t ws_size,
                              hipStream_t stream)
{
    const float* X   = (const float*)d_in[0];
    const int*   src = (const int*)  d_in[1];
    const int*   dst = (const int*)  d_in[2];
    const float* Wp1 = (const float*)d_in[3];
    const float* bp1 = (const float*)d_in[4];
    const float* Ws1 = (const float*)d_in[5];
    const float* bs1 = (const float*)d_in[6];
    const float* Wn1 = (const float*)d_in[7];
    const float* bn1 = (const float*)d_in[8];
    const float* Wp2 = (const float*)d_in[9];
    const float* bp2 = (const float*)d_in[10];
    const float* Ws2 = (const float*)d_in[11];
    const float* bs2 = (const float*)d_in[12];
    const float* Wn2 = (const float*)d_in[13];
    const float* bn2 = (const float*)d_in[14];

    const int N = in_sizes[0] / 128;
    const int E = in_sizes[1];

    float* pooled = (float*)d_ws;
    float* agg    = pooled + (size_t)N * 128;
    float* h1     = agg    + (size_t)N * 128;
    float* out    = (float*)d_out;

    const int tiles    = (N + 15) / 16;
    const int gemmGrid = (tiles + 3) / 4;               // 4 waves (M-tiles) per block
    const int scatGrid = (int)(((size_t)E * 32 + 255) / 256);
    const int n4       = N * 32;                        // N*128 floats as float4
    const int zeroGrid = 2048;

    // ---- layer 1 ----
    sage_gemm<8><<<gemmGrid, 128, 0, stream>>>(X, Wp1, bp1, nullptr, nullptr, nullptr,
                                               pooled, N, 1);
    zero_kernel<<<zeroGrid, 256, 0, stream>>>((float4*)agg, n4);
    sage_scatter_max<<<scatGrid, 256, 0, stream>>>(pooled, src, dst,
                                                   (unsigned int*)agg, E);
    sage_gemm<8><<<gemmGrid, 128, 0, stream>>>(X, Ws1, bs1, agg, Wn1, bn1,
                                               h1, N, 1);            // relu (layer-1 act)

    // ---- layer 2 ----
    sage_gemm<8><<<gemmGrid, 128, 0, stream>>>(h1, Wp2, bp2, nullptr, nullptr, nullptr,
                                               pooled, N, 1);
    zero_kernel<<<zeroGrid, 256, 0, stream>>>((float4*)agg, n4);
    sage_scatter_max<<<scatGrid, 256, 0, stream>>>(pooled, src, dst,
                                                   (unsigned int*)agg, E);
    sage_gemm<4><<<gemmGrid, 128, 0, stream>>>(h1, Ws2, bs2, agg, Wn2, bn2,
                                               out, N, 0);           // no final act
}